// NCModelForCoinFlip_74448963109338
// MI455X (gfx1250) — compile-verified
//
#include <hip/hip_runtime.h>

// out[:, :16384] = softmax(H_L, axis=1); out[:, 16384:] = beta * H_C
// Memory-bound: ~33 MB of traffic -> ~1.4 us floor at 23.3 TB/s.
// CDNA5 path: async global->LDS DMA (ASYNCcnt) for row staging, wave32 shuffles.

#define THREADS   512
#define WAVES     (THREADS / 32)
#define D_L       16384
#define D_C       64
#define D_OUT     (D_L + D_C)
#define V4T       (D_L / 4 / THREADS)   // float4 chunks per thread = 8

typedef __attribute__((address_space(3))) float lds_f;

__global__ __launch_bounds__(THREADS)
void fused_softmax_concat(const float* __restrict__ H_L,
                          const float* __restrict__ H_C,
                          const int*   __restrict__ beta_p,
                          float*       __restrict__ out)
{
    __shared__ float row[D_L];     // 64 KB staging buffer (4 WGs fit in 320 KB WGP LDS)
    float* red = row;              // reused for cross-wave reductions once row[] is consumed

    const int t    = threadIdx.x;
    const int lane = t & 31;       // wave32
    const int wave = t >> 5;
    const int r    = blockIdx.x;

    const float* g_row = H_L + (size_t)r * D_L;
    float*       o_row = out + (size_t)r * D_OUT;

    // ---- CDNA5 async global->LDS staging of one 64 KB row (ASYNCcnt path) ----
    {
        const unsigned lds_base = (unsigned)(unsigned long long)(lds_f*)row;
        #pragma unroll
        for (int i = 0; i < V4T; ++i) {
            const int idx4      = i * THREADS + t;          // coalesced float4 index
            const float* ga     = g_row + (size_t)idx4 * 4;
            const unsigned la   = lds_base + (unsigned)idx4 * 16u;
            asm volatile("global_load_async_to_lds_b128 %0, %1, off"
                         :: "v"(la), "v"(ga) : "memory");
        }
    }

    // Independent tail while the DMA is in flight: out[:, d_L:] = beta * H_C
    if (t < D_C) {
        const float betaf = (float)beta_p[0];
        o_row[D_L + t] = betaf * H_C[(size_t)r * D_C + t];
    }

    asm volatile("s_wait_asynccnt 0" ::: "memory");  // this wave's DMA done
    __syncthreads();                                 // all waves' portions visible

    // ---- pull 32 values/thread from LDS into VGPRs (single LDS read pass) ----
    float4 v[V4T];
    const float4* row4 = (const float4*)row;
    #pragma unroll
    for (int i = 0; i < V4T; ++i) v[i] = row4[i * THREADS + t];
    __syncthreads();   // row[] now dead -> safe to reuse as red[]

    // ---- max reduction ----
    float m = -__builtin_inff();
    #pragma unroll
    for (int i = 0; i < V4T; ++i)
        m = fmaxf(m, fmaxf(fmaxf(v[i].x, v[i].y), fmaxf(v[i].z, v[i].w)));
    #pragma unroll
    for (int off = 16; off > 0; off >>= 1) m = fmaxf(m, __shfl_xor(m, off));
    if (lane == 0) red[wave] = m;
    __syncthreads();
    float M = red[0];
    #pragma unroll
    for (int w = 1; w < WAVES; ++w) M = fmaxf(M, red[w]);
    __syncthreads();   // before red[] is reused for sums

    // ---- exp + sum reduction ----
    float s = 0.f;
    #pragma unroll
    for (int i = 0; i < V4T; ++i) {
        v[i].x = __expf(v[i].x - M); s += v[i].x;
        v[i].y = __expf(v[i].y - M); s += v[i].y;
        v[i].z = __expf(v[i].z - M); s += v[i].z;
        v[i].w = __expf(v[i].w - M); s += v[i].w;
    }
    #pragma unroll
    for (int off = 16; off > 0; off >>= 1) s += __shfl_xor(s, off);
    if (lane == 0) red[wave] = s;
    __syncthreads();
    float S = 0.f;
    #pragma unroll
    for (int w = 0; w < WAVES; ++w) S += red[w];
    const float inv = 1.0f / S;

    // ---- normalized store straight from registers (B128, coalesced) ----
    float4* o4 = (float4*)o_row;
    #pragma unroll
    for (int i = 0; i < V4T; ++i) {
        float4 e = v[i];
        e.x *= inv; e.y *= inv; e.z *= inv; e.w *= inv;
        o4[i * THREADS + t] = e;
    }
}

extern "C" void kernel_launch(void* const* d_in, const int* in_sizes, int n_in,
                              void* d_out, int out_size, void* d_ws, size_t ws_size,
                              hipStream_t stream) {
    const float* H_L  = (const float*)d_in[0];   // [B, 16384] f32
    const float* H_C  = (const float*)d_in[1];   // [B, 64]    f32
    const int*   beta = (const int*)d_in[2];     // scalar int
    float*       out  = (float*)d_out;           // [B, 16448] f32

    const int B = in_sizes[0] / D_L;             // 256
    fused_softmax_concat<<<dim3(B), dim3(THREADS), 0, stream>>>(H_L, H_C, beta, out);
}